// ST_Block_55044300865782
// MI455X (gfx1250) — compile-verified
//
#include <hip/hip_runtime.h>

// ---------------- problem constants ----------------
#define B_DIM   4
#define C_DIM   16
#define N_DIM   512
#define T_DIM   64
#define NT      (N_DIM * T_DIM)      // 32768
#define P_ORD   10
#define Q_ORD   5
#define R_CNT   66                   // (P+1)*(Q+1)
#define COUT    64
#define EPSV    1e-8f
#define BN_EPSV 1e-5f
#define SPS     516                  // padded LDS stride (bank-conflict-free)

typedef __attribute__((ext_vector_type(2))) float v2f;
typedef __attribute__((ext_vector_type(8))) float v8f;

#ifndef __has_builtin
#define __has_builtin(x) 0
#endif
#if __has_builtin(__builtin_amdgcn_global_load_async_to_lds_b32)
#define HAVE_ASYNC_LDS 1
#else
#define HAVE_ASYNC_LDS 0
#endif

#if HAVE_ASYNC_LDS
__device__ __forceinline__ void async_g2l_b32(const float* g, float* l) {
  __builtin_amdgcn_global_load_async_to_lds_b32(
      (__attribute__((address_space(1))) int*)(uintptr_t)g,
      (__attribute__((address_space(3))) int*)l, 0, 0);
}
__device__ __forceinline__ void wait_async() {
  asm volatile("s_wait_asynccnt 0x0" ::: "memory");
}
#endif

// ---------------- workspace layout (float offsets) ----------------
#define THETA_OFF   0
#define W2_OFF      512
#define B2_OFF      2560
#define POOL_OFF    4096
#define POOL_PER_WG (6 * NT)
#define XST_OFF     (POOL_OFF + 64 * POOL_PER_WG)

// =====================================================================
// Kernel A: tiny theta MLP + BN-folded output weights
// =====================================================================
__global__ void theta_fold_kernel(const float* __restrict__ STE,
                                  const float* __restrict__ w_t1,
                                  const float* __restrict__ b_t1,
                                  const float* __restrict__ w_t2,
                                  const float* __restrict__ b_t2,
                                  const float* __restrict__ w_mlp,
                                  const float* __restrict__ b_mlp,
                                  const float* __restrict__ gma,
                                  const float* __restrict__ bta,
                                  const float* __restrict__ mea,
                                  const float* __restrict__ var,
                                  float* __restrict__ ws) {
  const int tid = threadIdx.x;
  if (tid < B_DIM * R_CNT) {
    const int b = tid / R_CNT;
    const int r = tid % R_CNT;
    const int t = r / (P_ORD + 1);
    const int p = r % (P_ORD + 1);
    float acc = b_t2[p];
    for (int s = 0; s < P_ORD; ++s) {
      float h1 = b_t1[t];
      for (int u = 0; u < Q_ORD; ++u)
        h1 += w_t1[t * Q_ORD + u] * STE[(b * Q_ORD + u) * P_ORD + s];
      acc += w_t2[p * P_ORD + s] * h1;
    }
    ws[THETA_OFF + tid] = fmaxf(acc, 0.0f);
  }
  for (int i = tid; i < COUT * 2 * C_DIM; i += blockDim.x) {
    const int o = i / (2 * C_DIM);
    const float inv = gma[o] * rsqrtf(var[o] + BN_EPSV);
    ws[W2_OFF + i] = w_mlp[i] * inv;
  }
  if (tid < COUT) {
    const float inv = gma[tid] * rsqrtf(var[tid] + BN_EPSV);
    ws[B2_OFF + tid] = b_mlp[tid] * inv + bta[tid] - mea[tid] * inv;
  }
}

// =====================================================================
// deterministic block reduction (256 threads, 8 wave32)
// =====================================================================
__device__ __forceinline__ float block_reduce(float v, float* sred,
                                              float* sout, int slot) {
  const int lane = threadIdx.x & 31;
  const int wv = threadIdx.x >> 5;
#pragma unroll
  for (int o = 16; o > 0; o >>= 1) v += __shfl_down(v, o, 32);
  if (lane == 0) sred[wv] = v;
  __syncthreads();
  if (threadIdx.x == 0) {
    float t = 0.f;
#pragma unroll
    for (int i = 0; i < 8; ++i) t += sred[i];
    sout[slot] = t;
  }
  __syncthreads();
  return sout[slot];
}

// =====================================================================
// one 512-row GEMM: rst(512 x 64) = A(512 x kdim) @ Btrans-in-LDS
// B staged transposed: sBT[col*SPS + k]  -> fragments are single ds_load_b64
// =====================================================================
__device__ __forceinline__ void gemm_strips(const float* __restrict__ A,
                                            int lda, int kdim,
                                            const float* __restrict__ sBT,
                                            float* __restrict__ rst, int wv,
                                            int lrow, int lkh) {
  for (int strip = 0; strip < 4; ++strip) {
    const int rb = (strip * 8 + wv) * 16;
    const float* arow = A + (size_t)(rb + lrow) * lda;
    __builtin_prefetch(arow, 0, 1);
    v8f acc0 = {}, acc1 = {}, acc2 = {}, acc3 = {};
    for (int k0 = 0; k0 < kdim; k0 += 4) {
      const int ks = k0 + lkh * 2;
      v2f a = *(const v2f*)(arow + ks);
      v2f fb0 = *(const v2f*)&sBT[(lrow)*SPS + ks];
      v2f fb1 = *(const v2f*)&sBT[(16 + lrow) * SPS + ks];
      v2f fb2 = *(const v2f*)&sBT[(32 + lrow) * SPS + ks];
      v2f fb3 = *(const v2f*)&sBT[(48 + lrow) * SPS + ks];
      acc0 = __builtin_amdgcn_wmma_f32_16x16x4_f32(false, a, false, fb0, (short)0, acc0, false, false);
      acc1 = __builtin_amdgcn_wmma_f32_16x16x4_f32(false, a, false, fb1, (short)0, acc1, false, false);
      acc2 = __builtin_amdgcn_wmma_f32_16x16x4_f32(false, a, false, fb2, (short)0, acc2, false, false);
      acc3 = __builtin_amdgcn_wmma_f32_16x16x4_f32(false, a, false, fb3, (short)0, acc3, false, false);
    }
    const int orow = rb + 8 * lkh;
#pragma unroll
    for (int v = 0; v < 8; ++v) {
      rst[(orow + v) * T_DIM + lrow] = acc0[v];
      rst[(orow + v) * T_DIM + 16 + lrow] = acc1[v];
      rst[(orow + v) * T_DIM + 32 + lrow] = acc2[v];
      rst[(orow + v) * T_DIM + 48 + lrow] = acc3[v];
    }
  }
}

// =====================================================================
// Kernel B: per-(b,f) basis recursion. 64 blocks x 256 threads.
// =====================================================================
__global__ __launch_bounds__(256) void basis_kernel(
    const float* __restrict__ x, const float* __restrict__ Ls,
    const float* __restrict__ Lt, float* __restrict__ ws) {
  __shared__ float sBT[T_DIM * SPS];  // 129 KB transposed+padded B staging
  __shared__ float sred[8];
  __shared__ float sval[4];

  const int wg = blockIdx.x;  // b*16 + f
  const int b = wg >> 4;
  const int tid = threadIdx.x;
  const int lane = tid & 31;
  const int wv = tid >> 5;
  const int lrow = lane & 15;
  const int lkh = lane >> 4;

  const float* xs = x + (size_t)wg * NT;
  float* pool = ws + POOL_OFF + (size_t)wg * POOL_PER_WG;
  float* xst = ws + XST_OFF + (size_t)wg * NT;
  const float* theta = ws + THETA_OFF + b * R_CNT;

  // ---- m00 = x / max(||x||_F, eps); x_st = theta[0]*m00 ----
  float psq = 0.f;
  for (int e = tid; e < NT / 4; e += 256) {
    float4 v = ((const float4*)xs)[e];
    psq += v.x * v.x + v.y * v.y + v.z * v.z + v.w * v.w;
  }
  const float nrm0 = block_reduce(psq, sred, sval, 0);
  const float inv0 = 1.0f / fmaxf(sqrtf(nrm0), EPSV);
  const float th0 = theta[0];
  for (int e = tid; e < NT / 4; e += 256) {
    float4 v = ((const float4*)xs)[e];
    float4 m, a;
    m.x = v.x * inv0; m.y = v.y * inv0; m.z = v.z * inv0; m.w = v.w * inv0;
    a.x = th0 * m.x;  a.y = th0 * m.y;  a.z = th0 * m.z;  a.w = th0 * m.w;
    ((float4*)pool)[e] = m;
    ((float4*)xst)[e] = a;
  }
  __threadfence();
  __syncthreads();

  int last_s = 0, sec_s = -1, last_t = 0, sec_t = -1;

  for (int r = 1; r < R_CNT; ++r) {
    const bool spatial = (r % (Q_ORD + 1)) == 0;

    unsigned used = (1u << last_s) | (1u << last_t);
    if (sec_s >= 0) used |= 1u << sec_s;
    if (sec_t >= 0) used |= 1u << sec_t;
    const int dst = __builtin_ctz(~used & 0x3fu);
    float* rst = pool + (size_t)dst * NT;

    const int la = spatial ? last_s : last_t;
    const int se = spatial ? sec_s : sec_t;
    const float* lastBuf = pool + (size_t)la * NT;
    const float* secBuf = (se >= 0) ? (pool + (size_t)se * NT) : nullptr;

    if (spatial) {
      // stage last(512x64) transposed: sBT[t*SPS + m] = last[m*64 + t]
#if HAVE_ASYNC_LDS
      for (int e = tid; e < NT; e += 256) {
        const int m = e >> 6, t = e & 63;
        async_g2l_b32(lastBuf + e, &sBT[t * SPS + m]);
      }
      wait_async();
      __syncthreads();
#else
      for (int e = tid; e < NT / 4; e += 256) {
        const int m = e >> 4;
        const int t4 = (e & 15) * 4;
        float4 v = ((const float4*)lastBuf)[e];
        sBT[(t4 + 0) * SPS + m] = v.x;
        sBT[(t4 + 1) * SPS + m] = v.y;
        sBT[(t4 + 2) * SPS + m] = v.z;
        sBT[(t4 + 3) * SPS + m] = v.w;
      }
      __syncthreads();
#endif
      gemm_strips(Ls, N_DIM, N_DIM, sBT, rst, wv, lrow, lkh);
    } else {
      // stage L_t(64x64) transposed: sBT[u*SPS + t] = Lt[t*64 + u]
#if HAVE_ASYNC_LDS
      for (int e = tid; e < T_DIM * T_DIM; e += 256) {
        const int t = e >> 6, u = e & 63;
        async_g2l_b32(Lt + e, &sBT[u * SPS + t]);
      }
      wait_async();
      __syncthreads();
#else
      for (int e = tid; e < (T_DIM * T_DIM) / 4; e += 256) {
        const int t = e >> 4;
        const int u4 = (e & 15) * 4;
        float4 v = ((const float4*)Lt)[e];
        sBT[(u4 + 0) * SPS + t] = v.x;
        sBT[(u4 + 1) * SPS + t] = v.y;
        sBT[(u4 + 2) * SPS + t] = v.z;
        sBT[(u4 + 3) * SPS + t] = v.w;
      }
      __syncthreads();
#endif
      gemm_strips(lastBuf, T_DIM, T_DIM, sBT, rst, wv, lrow, lkh);
    }
    __threadfence();
    __syncthreads();

    // ---- Gram-Schmidt (matches reference's sequential projections) ----
    float d1p = 0.f, drsp = 0.f, dlsp = 0.f;
    for (int e = tid; e < NT / 4; e += 256) {
      const float4 rv = ((const float4*)rst)[e];
      const float4 lv = ((const float4*)lastBuf)[e];
      const float4 sv =
          secBuf ? ((const float4*)secBuf)[e] : make_float4(0.f, 0.f, 0.f, 0.f);
      d1p += rv.x * lv.x + rv.y * lv.y + rv.z * lv.z + rv.w * lv.w;
      drsp += rv.x * sv.x + rv.y * sv.y + rv.z * sv.z + rv.w * sv.w;
      dlsp += lv.x * sv.x + lv.y * sv.y + lv.z * sv.z + lv.w * sv.w;
    }
    const float d1 = block_reduce(d1p, sred, sval, 0);
    const float drs = block_reduce(drsp, sred, sval, 1);
    const float dls = block_reduce(dlsp, sred, sval, 2);
    const float d2 = drs - d1 * dls;  // = <rst - d1*last, sec>

    float sqp = 0.f;
    for (int e = tid; e < NT / 4; e += 256) {
      const float4 rv = ((const float4*)rst)[e];
      const float4 lv = ((const float4*)lastBuf)[e];
      const float4 sv =
          secBuf ? ((const float4*)secBuf)[e] : make_float4(0.f, 0.f, 0.f, 0.f);
      float4 o;
      o.x = rv.x - d1 * lv.x - d2 * sv.x;
      o.y = rv.y - d1 * lv.y - d2 * sv.y;
      o.z = rv.z - d1 * lv.z - d2 * sv.z;
      o.w = rv.w - d1 * lv.w - d2 * sv.w;
      ((float4*)rst)[e] = o;
      sqp += o.x * o.x + o.y * o.y + o.z * o.z + o.w * o.w;
    }
    const float ssq = block_reduce(sqp, sred, sval, 3);
    const float invn = 1.0f / fmaxf(sqrtf(ssq), EPSV);
    const float th = theta[r];
    for (int e = tid; e < NT / 4; e += 256) {
      float4 rv = ((const float4*)rst)[e];
      float4 xa = ((const float4*)xst)[e];
      rv.x *= invn; rv.y *= invn; rv.z *= invn; rv.w *= invn;
      xa.x += th * rv.x; xa.y += th * rv.y; xa.z += th * rv.z; xa.w += th * rv.w;
      ((float4*)rst)[e] = rv;
      ((float4*)xst)[e] = xa;
    }
    __threadfence();
    __syncthreads();

    if (spatial) {
      sec_s = last_s; last_s = dst;
      last_t = dst;   sec_t = -1;
    } else {
      sec_t = last_t; last_t = dst;
    }
  }
}

// =====================================================================
// Kernel C: out = BNfold(w_mlp) @ concat(x, x_st) + b'   (f32 WMMA)
// =====================================================================
__global__ __launch_bounds__(256) void out_kernel(const float* __restrict__ x,
                                                  const float* __restrict__ ws,
                                                  float* __restrict__ out) {
  __shared__ float sCT[32 * 36];  // [pos][k], padded stride 36

  const int b = blockIdx.y;
  const int pbase = blockIdx.x * 32;
  const int tid = threadIdx.x;
  const int lane = tid & 31;
  const int wv = tid >> 5;
  const int lrow = lane & 15;
  const int lkh = lane >> 4;
  const int mtile = wv & 3;   // 4 x 16 output rows
  const int ptile = wv >> 2;  // 2 x 16 positions

  const float* w2 = ws + W2_OFF;
  const float* b2 = ws + B2_OFF;
  const float* xb = x + (size_t)b * C_DIM * NT;
  const float* xstb = ws + XST_OFF + (size_t)b * C_DIM * NT;

  // stage B tile (32 channels x 32 positions), transposed
  for (int i = tid; i < 32 * 32; i += 256) {
    const int k = i >> 5, pos = i & 31;
    const float* src =
        (k < C_DIM) ? (xb + (size_t)k * NT) : (xstb + (size_t)(k - C_DIM) * NT);
#if HAVE_ASYNC_LDS
    async_g2l_b32(src + pbase + pos, &sCT[pos * 36 + k]);
#else
    sCT[pos * 36 + k] = src[pbase + pos];
#endif
  }
#if HAVE_ASYNC_LDS
  wait_async();
#endif
  __syncthreads();

  const int pos = pbase + ptile * 16 + lrow;
  const int colbase = (ptile * 16 + lrow) * 36;
  const int orow0 = mtile * 16 + lrow;

  v8f acc = {};
#pragma unroll
  for (int k0 = 0; k0 < 2 * C_DIM; k0 += 4) {
    const int ks = k0 + lkh * 2;
    v2f a = *(const v2f*)(w2 + orow0 * (2 * C_DIM) + ks);
    v2f bb = *(const v2f*)&sCT[colbase + ks];
    acc = __builtin_amdgcn_wmma_f32_16x16x4_f32(false, a, false, bb, (short)0, acc, false, false);
  }
  const int obase = mtile * 16 + 8 * lkh;
#pragma unroll
  for (int v = 0; v < 8; ++v) {
    const int o = obase + v;
    out[((size_t)(b * COUT + o)) * NT + pos] = acc[v] + b2[o];
  }
}

// =====================================================================
extern "C" void kernel_launch(void* const* d_in, const int* in_sizes, int n_in,
                              void* d_out, int out_size, void* d_ws,
                              size_t ws_size, hipStream_t stream) {
  const float* x = (const float*)d_in[0];
  const float* Ls = (const float*)d_in[1];
  const float* Lt = (const float*)d_in[2];
  const float* STE = (const float*)d_in[3];
  const float* w_t1 = (const float*)d_in[4];
  const float* b_t1 = (const float*)d_in[5];
  const float* w_t2 = (const float*)d_in[6];
  const float* b_t2 = (const float*)d_in[7];
  const float* w_mlp = (const float*)d_in[8];
  const float* b_mlp = (const float*)d_in[9];
  const float* gma = (const float*)d_in[10];
  const float* bta = (const float*)d_in[11];
  const float* mea = (const float*)d_in[12];
  const float* var = (const float*)d_in[13];
  float* ws = (float*)d_ws;
  float* out = (float*)d_out;

  theta_fold_kernel<<<dim3(1), dim3(320), 0, stream>>>(
      STE, w_t1, b_t1, w_t2, b_t2, w_mlp, b_mlp, gma, bta, mea, var, ws);
  basis_kernel<<<dim3(B_DIM * C_DIM), dim3(256), 0, stream>>>(x, Ls, Lt, ws);
  out_kernel<<<dim3(NT / 32, B_DIM), dim3(256), 0, stream>>>(x, ws, out);
}